// Edgewise_GATLayer_19868518711924
// MI455X (gfx1250) — compile-verified
//
#include <hip/hip_runtime.h>
#include <math.h>

#define NN   50000
#define EE   800000
#define BB   2
#define IND  64
#define OUTD 32
#define NBO  (NN * BB * OUTD)   /* 3,200,000 elements */

typedef __attribute__((ext_vector_type(2))) float v2f;
typedef __attribute__((ext_vector_type(8))) float v8f;

// ---------------------------------------------------------------------------
// Kernel 0: initialize m = -inf, denom = 0, out = 0
// ---------------------------------------------------------------------------
__global__ void init_kernel(float* __restrict__ m, float* __restrict__ denom,
                            float* __restrict__ out) {
    int i = blockIdx.x * blockDim.x + threadIdx.x;
    if (i < NBO) {
        m[i]     = -INFINITY;
        denom[i] = 0.0f;
        out[i]   = 0.0f;
    }
}

// ---------------------------------------------------------------------------
// Kernel 1: z = h @ W_fc + b_fc via V_WMMA_F32_16X16X4_F32.
// One wave computes a 16(node) x 16(out) f32 tile for one batch; K=64 is a
// chain of 16 WMMAs. Fragment layouts per CDNA5 ISA 7.12.2:
//   A (16x4 f32): lanes 0-15 hold row M=lane, K={k,k+1}; lanes 16-31 hold
//                 row M=lane-16, K={k+2,k+3}.
//   B (4x16 f32): lanes 0-15 hold col N=lane, K={k,k+1}; lanes 16-31 hold
//                 col N=lane-16, K={k+2,k+3}.
//   C/D: VGPR j -> M = j + 8*(lane>=16), N = lane&15.
// z is written node-major: z[n, b, o] at n*(B*OUT) + b*OUT + o.
// ---------------------------------------------------------------------------
__global__ __launch_bounds__(256) void fc_wmma_kernel(
    const float* __restrict__ h, const float* __restrict__ W,
    const float* __restrict__ bfc, float* __restrict__ z) {
    const int NWAVES = (NN / 16) * BB * (OUTD / 16);   // 12500
    int wid = (blockIdx.x * blockDim.x + threadIdx.x) >> 5;  // wave-uniform
    if (wid >= NWAVES) return;                                // whole-wave exit

    int lane = threadIdx.x & 31;
    int half = lane >> 4;     // 0: K pair {0,1}; 1: K pair {2,3}
    int lm   = lane & 15;     // M for A, N for B

    int oc    = wid & 1;          // out-column tile (0 or 1)
    int b     = (wid >> 1) & 1;   // batch
    int tile  = wid >> 2;         // node tile
    int node0 = tile * 16;
    int out0  = oc * 16;

    const float* Arow = h + ((size_t)b * NN + (size_t)(node0 + lm)) * IND;

    v8f acc = {};
    #pragma unroll
    for (int k = 0; k < IND; k += 4) {
        v2f a, bm;
        int ka = k + 2 * half;
        a.x  = Arow[ka + 0];
        a.y  = Arow[ka + 1];
        bm.x = W[(ka + 0) * OUTD + out0 + lm];
        bm.y = W[(ka + 1) * OUTD + out0 + lm];
        acc = __builtin_amdgcn_wmma_f32_16x16x4_f32(
            /*neg_a=*/false, a, /*neg_b=*/false, bm,
            /*c_mod=*/(short)0, acc, /*reuse_a=*/false, /*reuse_b=*/false);
    }

    float bias = bfc[out0 + lm];
    #pragma unroll
    for (int j = 0; j < 8; ++j) {
        int row = j + 8 * half;
        z[(size_t)(node0 + row) * (BB * OUTD) + b * OUTD + out0 + lm] =
            acc[j] + bias;
    }
}

// ---------------------------------------------------------------------------
// Float atomic max via int/uint atomics (valid for mixed signs w/ -inf init).
// ---------------------------------------------------------------------------
__device__ __forceinline__ void atomic_max_f32(float* addr, float v) {
    if (v >= 0.0f)
        atomicMax((int*)addr, __float_as_int(v));
    else
        atomicMin((unsigned int*)addr, (unsigned int)__float_as_int(v));
}

// ---------------------------------------------------------------------------
// Kernel 2: segment max of gathered src features over dst.
// Thread = (edge, batch, out-channel); 32 consecutive threads read one
// 128B z-row (L2-resident, 12.8 MB).
// ---------------------------------------------------------------------------
__global__ void segmax_kernel(const float* __restrict__ z,
                              const int* __restrict__ src,
                              const int* __restrict__ dst,
                              float* __restrict__ m) {
    int tid = blockIdx.x * blockDim.x + threadIdx.x;
    if (tid >= EE * BB * OUTD) return;
    int o = tid & 31;
    int b = (tid >> 5) & 1;
    int e = tid >> 6;
    int s = src[e], d = dst[e];
    float szv = z[(size_t)s * (BB * OUTD) + b * OUTD + o];
    atomic_max_f32(&m[(size_t)d * (BB * OUTD) + b * OUTD + o], szv);
}

// ---------------------------------------------------------------------------
// Kernel 3: denom = segment_sum(exp(sz - m[dst]))
// ---------------------------------------------------------------------------
__global__ void denom_kernel(const float* __restrict__ z,
                             const int* __restrict__ src,
                             const int* __restrict__ dst,
                             const float* __restrict__ m,
                             float* __restrict__ denom) {
    int tid = blockIdx.x * blockDim.x + threadIdx.x;
    if (tid >= EE * BB * OUTD) return;
    int o = tid & 31;
    int b = (tid >> 5) & 1;
    int e = tid >> 6;
    int s = src[e], d = dst[e];
    size_t di = (size_t)d * (BB * OUTD) + b * OUTD + o;
    float ex = expf(z[(size_t)s * (BB * OUTD) + b * OUTD + o] - m[di]);
    atomicAdd(&denom[di], ex);
}

// ---------------------------------------------------------------------------
// Kernel 4: per-(edge,batch) wave computes the attention logit via a wave32
// shuffle reduction, then scatters alpha * e into out[b, dst, o].
// ---------------------------------------------------------------------------
__global__ void edge_out_kernel(const float* __restrict__ z,
                                const float* __restrict__ weight,
                                const int* __restrict__ src,
                                const int* __restrict__ dst,
                                const float* __restrict__ Wa,
                                const float* __restrict__ ba,
                                const float* __restrict__ m,
                                const float* __restrict__ denom,
                                float* __restrict__ out) {
    int tid = blockIdx.x * blockDim.x + threadIdx.x;
    int wid = tid >> 5;
    if (wid >= EE * BB) return;
    int lane = tid & 31;
    int b = wid & 1;
    int e = wid >> 1;
    int s = src[e], d = dst[e];

    size_t si = (size_t)s * (BB * OUTD) + b * OUTD + lane;
    size_t di = (size_t)d * (BB * OUTD) + b * OUTD + lane;
    float szv = z[si];
    float dzv = z[di];

    // att logit: cat[sz, dz, w] @ W_att + b_att  (wave32 reduction over 32 ch)
    float part = szv * Wa[lane] + dzv * Wa[OUTD + lane];
    #pragma unroll
    for (int off = 16; off; off >>= 1)
        part += __shfl_xor(part, off, 32);
    float logit = part + weight[e] * Wa[2 * OUTD] + ba[0];
    float ev = (logit >= 0.0f) ? logit : 0.01f * logit;   // leaky_relu

    // per-feature softmax weight (faithful to the reference quirk)
    float ex    = expf(szv - m[di]);
    float alpha = ex / denom[di];

    atomicAdd(&out[(size_t)b * NN * OUTD + (size_t)d * OUTD + lane],
              alpha * ev);
}

// ---------------------------------------------------------------------------
extern "C" void kernel_launch(void* const* d_in, const int* in_sizes, int n_in,
                              void* d_out, int out_size, void* d_ws, size_t ws_size,
                              hipStream_t stream) {
    const float* h      = (const float*)d_in[0];
    const float* weight = (const float*)d_in[1];
    const int*   src    = (const int*)d_in[2];
    const int*   dst    = (const int*)d_in[3];
    const float* W_fc   = (const float*)d_in[4];
    const float* b_fc   = (const float*)d_in[5];
    const float* W_att  = (const float*)d_in[6];
    const float* b_att  = (const float*)d_in[7];
    float* out = (float*)d_out;

    float* z     = (float*)d_ws;     // [N, B, OUT]  12.8 MB
    float* m     = z + NBO;          // [N, B, OUT]  12.8 MB
    float* denom = m + NBO;          // [N, B, OUT]  12.8 MB

    // 0) init m / denom / out
    init_kernel<<<(NBO + 255) / 256, 256, 0, stream>>>(m, denom, out);

    // 1) FC via WMMA f32 16x16x4
    const int NWAVES = (NN / 16) * BB * (OUTD / 16);      // 12500 waves
    fc_wmma_kernel<<<(NWAVES * 32 + 255) / 256, 256, 0, stream>>>(h, W_fc, b_fc, z);

    // 2) segment max
    const int ETH = EE * BB * OUTD;                        // 51.2M threads
    segmax_kernel<<<(ETH + 255) / 256, 256, 0, stream>>>(z, src, dst, m);

    // 3) denom = segment_sum(exp(sz - m))
    denom_kernel<<<(ETH + 255) / 256, 256, 0, stream>>>(z, src, dst, m, denom);

    // 4) attention + weighted scatter into out[B,N,OUT]
    edge_out_kernel<<<(ETH + 255) / 256, 256, 0, stream>>>(
        z, weight, src, dst, W_att, b_att, m, denom, out);
}